// GNNLayer_30502857736676
// MI455X (gfx1250) — compile-verified
//
#include <hip/hip_runtime.h>

typedef __attribute__((ext_vector_type(2))) float v2f;
typedef __attribute__((ext_vector_type(8))) float v8f;

#define DIM 64

// ---------------- Phase 1: degree accumulation (2M float atomics) ----------
__global__ void deg_kernel(const int* __restrict__ row, float* __restrict__ deg, int E) {
    int e = blockIdx.x * blockDim.x + threadIdx.x;
    if (e < E) unsafeAtomicAdd(&deg[row[e]], 1.0f);
}

// ---------------- Phase 2: dinv = rsqrt(deg) in place ----------------------
__global__ void dinv_kernel(float* __restrict__ deg, int n) {
    int i = blockIdx.x * blockDim.x + threadIdx.x;
    if (i < n) {
        float d = deg[i];
        deg[i] = (d > 0.0f) ? __frsqrt_rn(d) : 0.0f;
    }
}

// ---------------- Phase 3: edge-parallel SpMM scatter ----------------------
// One 32-lane wave per edge; lane handles dims {lane, lane+32}.
// acc1 += v * embed[col], acc2 += v * embed[col]^2, scattered to row.
__global__ void __launch_bounds__(256) spmm_kernel(
    const int* __restrict__ row, const int* __restrict__ col,
    const float* __restrict__ dinv, const float* __restrict__ embed,
    float* __restrict__ acc1, float* __restrict__ acc2, int E) {
    int lane = threadIdx.x & 31;
    int e = (int)((blockIdx.x * blockDim.x + threadIdx.x) >> 5);
    if (e >= E) return;
    int r = row[e];
    int c = col[e];
    float v = dinv[r] * dinv[c];
    const float* xc = embed + (size_t)c * DIM;
    float x0 = xc[lane];
    float x1 = xc[lane + 32];
    size_t o = (size_t)r * DIM;
    unsafeAtomicAdd(&acc1[o + lane],      v * x0);
    unsafeAtomicAdd(&acc1[o + lane + 32], v * x1);
    unsafeAtomicAdd(&acc2[o + lane],      v * x0 * x0);
    unsafeAtomicAdd(&acc2[o + lane + 32], v * x1 * x1);
}

// ---------------- Phase 4: fused dual GEMM via V_WMMA_F32_16X16X4_F32 ------
// out = (embed + acc1) @ W1^T + acc2 @ W2^T + (b1 + b2)
// One wave -> one 16x16 output tile. 4 waves/block cover all 64 output cols
// of one 16-row slab. K=64 -> 16 WMMA steps per matrix, same accumulator.
__global__ void __launch_bounds__(128) gemm_kernel(
    const float* __restrict__ embed, const float* __restrict__ acc1,
    const float* __restrict__ acc2,
    const float* __restrict__ W1, const float* __restrict__ b1,
    const float* __restrict__ W2, const float* __restrict__ b2,
    float* __restrict__ out, int n) {
    const int lane = threadIdx.x & 31;
    const int wave = threadIdx.x >> 5;
    const int lo   = lane & 15;          // A: M index; B/C: N index
    const int hi   = lane >> 4;          // selects K pair (A/B) and M half (C)
    const int n0   = wave * 16;          // output-column tile of this wave
    const int ncol = n0 + lo;

    // Preload B fragments: B[k][n] = W[n][k]  (x @ W^T)
    // Layout: VGPR0 holds K=2*hi, VGPR1 holds K=2*hi+1 for column ncol.
    v2f B1[16], B2[16];
#pragma unroll
    for (int ks = 0; ks < 16; ++ks) {
        int k = 4 * ks + 2 * hi;
        B1[ks].x = W1[ncol * 64 + k];
        B1[ks].y = W1[ncol * 64 + k + 1];
        B2[ks].x = W2[ncol * 64 + k];
        B2[ks].y = W2[ncol * 64 + k + 1];
    }
    float bias = b1[ncol] + b2[ncol];

    int m0 = blockIdx.x * 16;            // row slab
    int rowA = m0 + lo;                  // A-fragment row for this lane
    if (rowA >= n) rowA = n - 1;         // clamp loads; keep EXEC all-ones
    size_t base = (size_t)rowA * 64 + 2 * hi;

    v8f c = {};
    // Pass 1: A = embed + acc1, multiply by W1^T
#pragma unroll
    for (int ks = 0; ks < 16; ++ks) {
        size_t idx = base + 4 * ks;
        v2f a;
        a.x = embed[idx]     + acc1[idx];
        a.y = embed[idx + 1] + acc1[idx + 1];
        c = __builtin_amdgcn_wmma_f32_16x16x4_f32(
                false, a, false, B1[ks], (short)0, c, false, false);
    }
    // Pass 2: A = acc2, multiply by W2^T, same accumulator
#pragma unroll
    for (int ks = 0; ks < 16; ++ks) {
        size_t idx = base + 4 * ks;
        v2f a;
        a.x = acc2[idx];
        a.y = acc2[idx + 1];
        c = __builtin_amdgcn_wmma_f32_16x16x4_f32(
                false, a, false, B2[ks], (short)0, c, false, false);
    }
    // C layout: VGPR r -> row m0 + r + 8*hi, col ncol. Predicate ragged rows.
#pragma unroll
    for (int r = 0; r < 8; ++r) {
        int mm = m0 + r + 8 * hi;
        if (mm < n) out[(size_t)mm * 64 + ncol] = c[r] + bias;
    }
}

extern "C" void kernel_launch(void* const* d_in, const int* in_sizes, int n_in,
                              void* d_out, int out_size, void* d_ws, size_t ws_size,
                              hipStream_t stream) {
    const float* embed = (const float*)d_in[0];
    const int*   row   = (const int*)d_in[1];
    const int*   col   = (const int*)d_in[2];
    const float* W1    = (const float*)d_in[3];
    const float* b1    = (const float*)d_in[4];
    const float* W2    = (const float*)d_in[5];
    const float* b2    = (const float*)d_in[6];
    float*       out   = (float*)d_out;

    const int n = in_sizes[0] / DIM;   // 144242 nodes
    const int E = in_sizes[1];         // 2,000,000 directed edges

    // Workspace: [deg/dinv : nPad floats][acc1 : n*64][acc2 : n*64]
    float* ws = (float*)d_ws;
    size_t nPad = ((size_t)n + 255) & ~(size_t)255;
    float* deg  = ws;
    float* acc1 = ws + nPad;
    float* acc2 = acc1 + (size_t)n * DIM;
    size_t zeroFloats = nPad + 2 * (size_t)n * DIM;

    hipMemsetAsync(d_ws, 0, zeroFloats * sizeof(float), stream);

    deg_kernel<<<(E + 255) / 256, 256, 0, stream>>>(row, deg, E);
    dinv_kernel<<<(n + 255) / 256, 256, 0, stream>>>(deg, n);

    // 8 edges per 256-thread block (one wave per edge)
    spmm_kernel<<<(E + 7) / 8, 256, 0, stream>>>(row, col, deg, embed, acc1, acc2, E);

    // one block per 16-row slab, 4 waves cover the 64 output columns
    gemm_kernel<<<(n + 15) / 16, 128, 0, stream>>>(embed, acc1, acc2,
                                                   W1, b1, W2, b2, out, n);
}